// GNN_9371618640103
// MI455X (gfx1250) — compile-verified
//
#include <hip/hip_runtime.h>

typedef float v2f __attribute__((ext_vector_type(2)));
typedef float v8f __attribute__((ext_vector_type(8)));

#define N_NODES 100000
#define N_EDGES 1600000
#define IN_DIM  128
#define HID     64   // HID_DIM == OUT_DIM == 64

// ---------------------------------------------------------------- degree/norm
__global__ void gcn_deg_init(float* __restrict__ deg) {
    int i = blockIdx.x * blockDim.x + threadIdx.x;
    if (i < N_NODES) deg[i] = 1.0f;   // self-loop pre-counted
}

__global__ void gcn_deg_count(const long long* __restrict__ ei,
                              float* __restrict__ deg) {
    int e = blockIdx.x * blockDim.x + threadIdx.x;
    if (e < N_EDGES) {
        int d = (int)ei[N_EDGES + e];   // dst row of edge_index
        atomicAdd(&deg[d], 1.0f);
    }
}

__global__ void gcn_dinv(float* __restrict__ deg) {
    int i = blockIdx.x * blockDim.x + threadIdx.x;
    if (i < N_NODES) deg[i] = rsqrtf(deg[i]);   // deg >= 1 always
}

// ---------------------------------------------------------------- WMMA GEMM
// C[100000 x 64] = A[100000 x K] @ B[K x 64], f32 via V_WMMA_F32_16X16X4_F32.
// One wave -> 32 rows x 64 cols (8 accumulator tiles). B staged in LDS,
// pre-paired along K so each B fragment is a single aligned ds_load_b64:
//   Bs2[p*64 + n] = { B[2p][n], B[2p+1][n] }
// Fused epilogue: lin = A@B ; agg = lin * dinv[row]^2 (+ bias[col] if given).
template <int K>
__global__ __launch_bounds__(128)
void gcn_gemm_wmma(const float* __restrict__ A, const float* __restrict__ B,
                   const float* __restrict__ dinv, const float* __restrict__ bias,
                   float* __restrict__ lin, float* __restrict__ agg) {
    __shared__ v2f Bs2[(K / 2) * 64];
    float* Bsf = (float*)Bs2;
    const int tid = threadIdx.x;
    for (int i = tid; i < K * 64; i += 128) {
        const int k = i >> 6;
        const int n = i & 63;
        Bsf[(((k >> 1) * 64) + n) * 2 + (k & 1)] = B[i];
    }
    __syncthreads();

    const int wave  = tid >> 5;
    const int lane  = tid & 31;
    const int strip = blockIdx.x * 4 + wave;
    if (strip >= N_NODES / 32) return;          // wave-uniform exit, EXEC stays full

    const int row0 = strip * 32;
    const int r    = lane & 15;                 // A row within tile / B-D column
    const int hi   = lane >> 4;
    const int kb   = hi * 2;                    // K sub-offset per ISA A/B layout

    const float* arow0 = A + (size_t)(row0 + r) * K + kb;
    const float* arow1 = A + (size_t)(row0 + 16 + r) * K + kb;

    v8f acc[8];
#pragma unroll
    for (int t = 0; t < 8; ++t) acc[t] = (v8f){};

    for (int k0 = 0; k0 < K; k0 += 4) {
        v2f a0 = *(const v2f*)(arow0 + k0);     // A[row0+r   ][k0+kb .. +1]
        v2f a1 = *(const v2f*)(arow1 + k0);     // A[row0+16+r][k0+kb .. +1]
        const v2f* bp = &Bs2[(k0 / 2 + hi) * 64 + r];
#pragma unroll
        for (int t = 0; t < 4; ++t) {
            v2f b = bp[t * 16];                 // {B[k0+kb][col], B[k0+kb+1][col]}
            acc[t] = __builtin_amdgcn_wmma_f32_16x16x4_f32(
                false, a0, false, b, (short)0, acc[t], false, false);
            acc[t + 4] = __builtin_amdgcn_wmma_f32_16x16x4_f32(
                false, a1, false, b, (short)0, acc[t + 4], false, false);
        }
    }

#pragma unroll
    for (int half = 0; half < 2; ++half) {
#pragma unroll
        for (int v = 0; v < 8; ++v) {
            const int row = row0 + half * 16 + v + 8 * hi;  // M = v + 8*(lane>>4)
            const float w = dinv[row];
            const float w2 = w * w;
#pragma unroll
            for (int t = 0; t < 4; ++t) {
                const int col = t * 16 + r;
                const float val = acc[half * 4 + t][v];
                const size_t idx = (size_t)row * 64 + col;
                lin[idx] = val;
                float o = val * w2;             // self-loop contribution
                if (bias) o += bias[col];
                agg[idx] = o;
            }
        }
    }
}

// ---------------------------------------------------------------- edge scatter
// 32 lanes per edge, 2 features per lane: agg[dst] += lin[src] * dinv[s]*dinv[d]
__global__ void gcn_scatter(const long long* __restrict__ ei,
                            const float* __restrict__ dinv,
                            const float* __restrict__ lin,
                            float* __restrict__ agg) {
    const long long tid = (long long)blockIdx.x * blockDim.x + threadIdx.x;
    const int e = (int)(tid >> 5);
    if (e >= N_EDGES) return;
    const int f = ((int)tid & 31) * 2;
    const int s = (int)ei[e];
    const int d = (int)ei[N_EDGES + e];
    const float norm = dinv[s] * dinv[d];
    v2f hv = *(const v2f*)(lin + (size_t)s * 64 + f);
    float* dst = agg + (size_t)d * 64 + f;
    atomicAdd(dst,     hv.x * norm);
    atomicAdd(dst + 1, hv.y * norm);
}

// ---------------------------------------------------------------- relu + bias
__global__ void gcn_relu_bias(float* __restrict__ h, const float* __restrict__ b) {
    int i = blockIdx.x * blockDim.x + threadIdx.x;
    if (i < N_NODES * 64) {
        float v = h[i] + b[i & 63];
        h[i] = v > 0.0f ? v : 0.0f;
    }
}

// ---------------------------------------------------------------- launch
extern "C" void kernel_launch(void* const* d_in, const int* in_sizes, int n_in,
                              void* d_out, int out_size, void* d_ws, size_t ws_size,
                              hipStream_t stream) {
    const float*     x  = (const float*)d_in[0];
    const long long* ei = (const long long*)d_in[1];   // int64, shape (2, E) flat
    const float*     W1 = (const float*)d_in[2];
    const float*     b1 = (const float*)d_in[3];
    const float*     W2 = (const float*)d_in[4];
    const float*     b2 = (const float*)d_in[5];
    float* out = (float*)d_out;

    // workspace: dinv[100000] | bufA (lin, 6.4M f32) | bufB (agg/h, 6.4M f32)
    char* ws = (char*)d_ws;
    float* dinv = (float*)ws;
    float* bufA = (float*)(ws + ((N_NODES * 4 + 255) / 256) * 256);
    float* bufB = bufA + (size_t)N_NODES * 64;

    const int NB_NODE = (N_NODES + 255) / 256;          // 391
    const int NB_EDGE = (N_EDGES + 255) / 256;          // 6250
    const int NB_SCAT = (N_EDGES * 32) / 256;           // 200000 (exact)
    const int NB_FEAT = (N_NODES * 64) / 256;           // 25000 (exact)
    const int NB_GEMM = (N_NODES / 32 + 3) / 4;         // 782

    // normalization
    gcn_deg_init <<<NB_NODE, 256, 0, stream>>>(dinv);
    gcn_deg_count<<<NB_EDGE, 256, 0, stream>>>(ei, dinv);
    gcn_dinv     <<<NB_NODE, 256, 0, stream>>>(dinv);

    // layer 1: lin1 = x@W1 -> bufA ; agg1 init = lin1*dinv^2 -> bufB
    gcn_gemm_wmma<IN_DIM><<<NB_GEMM, 128, 0, stream>>>(x, W1, dinv, nullptr, bufA, bufB);
    gcn_scatter  <<<NB_SCAT, 256, 0, stream>>>(ei, dinv, bufA, bufB);
    gcn_relu_bias<<<NB_FEAT, 256, 0, stream>>>(bufB, b1);

    // layer 2: lin2 = h@W2 -> bufA ; out init = lin2*dinv^2 + b2 -> d_out
    gcn_gemm_wmma<HID><<<NB_GEMM, 128, 0, stream>>>(bufB, W2, dinv, b2, bufA, out);
    gcn_scatter  <<<NB_SCAT, 256, 0, stream>>>(ei, dinv, bufA, out);
}